// HierarchicalGraphConv_64922725646518
// MI455X (gfx1250) — compile-verified
//
#include <hip/hip_runtime.h>
#include <hip/hip_bf16.h>
#include <math.h>

typedef float v2f __attribute__((ext_vector_type(2)));
typedef float v8f __attribute__((ext_vector_type(8)));

#define ND 128
#define ED 64
#define HD 256
#define NH 8
#define DH 32
#define LN_EPS 1e-5f

#define KMAX 256
#define KPAD 4                 // LDS row-stride pad: (K+4) % 64 == 4 -> conflict-free fragment reads

// ---------------------------------------------------------------------------
// Monotonic float<->uint encoding for atomic float max
// ---------------------------------------------------------------------------
__device__ __forceinline__ unsigned fenc(float f) {
    unsigned u = __float_as_uint(f);
    return (u & 0x80000000u) ? ~u : (u | 0x80000000u);
}
__device__ __forceinline__ float fdec(unsigned u) {
    unsigned v = (u & 0x80000000u) ? (u & 0x7FFFFFFFu) : ~u;
    return __uint_as_float(v);
}

// ---------------------------------------------------------------------------
// Generic fp32 WMMA GEMM:  C[M,N] = A[M,K] @ B[K,N] + bias[N]
// One wave computes a 16x16 tile via V_WMMA_F32_16X16X4_F32; 4 waves/block
// cover a 16x64 strip. The 16xK A-strip (contiguous, row-major) is staged
// once per block into LDS with coalesced float4 loads; row stride K+4 keeps
// the 16-lane b64 fragment reads bank-conflict-free. Inner loop is branch-
// free: OOB rows/cols are clamped (they only pollute D rows/cols that the
// guarded store never writes). EXEC is all-1s at every WMMA.
// ---------------------------------------------------------------------------
__global__ void wmma_gemm_bias(const float* __restrict__ A,
                               const float* __restrict__ B,
                               const float* __restrict__ bias,
                               float* __restrict__ C,
                               int M, int K, int N) {
    __shared__ float ldsA[16 * (KMAX + KPAD)];

    const int tid  = threadIdx.x;
    const int lane = tid & 31;
    const int wave = tid >> 5;
    const int row0 = blockIdx.x * 16;
    const int col0 = blockIdx.y * 64 + wave * 16;
    const int Kp   = K + KPAD;

    // ---- stage 16xK A strip into LDS (coalesced float4), zero-fill OOB rows
    {
        const float* __restrict__ strip = A + (long long)row0 * K;
        const int mrows = (M - row0) < 16 ? (M - row0) : 16;
        for (int r2 = 0; r2 < 16; ++r2) {
            const bool ok = r2 < mrows;
            for (int c = tid * 4; c < K; c += 128 * 4) {
                float4 vdat = ok ? *(const float4*)(strip + r2 * K + c)
                                 : make_float4(0.f, 0.f, 0.f, 0.f);
                *(float4*)(&ldsA[r2 * Kp + c]) = vdat;
            }
        }
    }
    __syncthreads();

    const int r = lane & 15;      // A-row / B-col within tile
    const int h = lane >> 4;      // K-half selector (0 or 1)
    const int col   = col0 + r;
    const bool colOK = col < N;
    const int bcol  = colOK ? col : 0;       // clamped: OOB column never stored

    v8f acc;
    const float bv = colOK ? bias[bcol] : 0.0f;
#pragma unroll
    for (int i = 0; i < 8; ++i) acc[i] = bv;

    const float* __restrict__ arow = ldsA + r * Kp + 2 * h;   // LDS fragment base
    const float* __restrict__ bp   = B + bcol;

    int boff = 2 * h * N;
    for (int k0 = 0; k0 < K; k0 += 4) {
        v2f a = *(const v2f*)(arow + k0);     // ds_load_b64, conflict-free
        v2f b;
        b[0] = bp[boff];
        b[1] = bp[boff + N];
        boff += 4 * N;
        acc = __builtin_amdgcn_wmma_f32_16x16x4_f32(
            /*neg_a=*/false, a, /*neg_b=*/false, b,
            /*c_mod=*/(short)0, acc, /*reuse_a=*/false, /*reuse_b=*/false);
    }

    if (colOK) {
#pragma unroll
        for (int i = 0; i < 8; ++i) {
            const int rr = row0 + h * 8 + i;
            if (rr < M) C[rr * N + col] = acc[i];
        }
    }
}

// ---------------------------------------------------------------------------
// Fill kernel (u32 pattern) for clearing m / denom / agg buffers
// ---------------------------------------------------------------------------
__global__ void fill_u32(unsigned* __restrict__ p, unsigned v, int n) {
    const int i = blockIdx.x * blockDim.x + threadIdx.x;
    if (i < n) p[i] = v;
}

// ---------------------------------------------------------------------------
// Per-edge multi-head scores:
//   s[e,h] = dot(q[src,h,:], k[tgt,h,:]) / sqrt(DH) + edge_att[e,h]
// then atomic segment-max into m[tgt,h] (uint encoded).
// ---------------------------------------------------------------------------
__global__ void scores_kernel(const float* __restrict__ q,
                              const float* __restrict__ k,
                              const int* __restrict__ ei,
                              float* __restrict__ sc,     // in: edge_att, out: scores
                              unsigned* __restrict__ mbuf,
                              int E) {
    const int idx = blockIdx.x * blockDim.x + threadIdx.x;
    if (idx >= E * NH) return;
    const int e = idx >> 3;
    const int h = idx & 7;
    const int src = ei[e];
    const int tgt = ei[E + e];
    const float4* __restrict__ qp = (const float4*)(q + (long long)src * HD + h * DH);
    const float4* __restrict__ kp = (const float4*)(k + (long long)tgt * HD + h * DH);
    float s = 0.0f;
#pragma unroll
    for (int t = 0; t < DH / 4; ++t) {
        const float4 a = qp[t];
        const float4 b = kp[t];
        s += a.x * b.x + a.y * b.y + a.z * b.z + a.w * b.w;
    }
    s = s * 0.17677669529663687f + sc[idx];   // 1/sqrt(32)
    sc[idx] = s;
    atomicMax(&mbuf[(long long)tgt * NH + h], fenc(s));
}

// ---------------------------------------------------------------------------
// ex = exp(s - m[tgt]);  denom[tgt,h] += ex  (stable segment softmax, pass 2)
// ---------------------------------------------------------------------------
__global__ void exp_kernel(float* __restrict__ sc,
                           const unsigned* __restrict__ mbuf,
                           float* __restrict__ denom,
                           const int* __restrict__ ei,
                           int E) {
    const int idx = blockIdx.x * blockDim.x + threadIdx.x;
    if (idx >= E * NH) return;
    const int e = idx >> 3;
    const int h = idx & 7;
    const int tgt = ei[E + e];
    const float m = fdec(mbuf[(long long)tgt * NH + h]);
    const float ex = expf(sc[idx] - m);
    sc[idx] = ex;
    atomicAdd(&denom[(long long)tgt * NH + h], ex);
}

// ---------------------------------------------------------------------------
// Weighted scatter aggregation: agg[tgt, :] += (ex/denom[tgt]) * v[tgt, :]
// (reference indexes v by tgt). One block of 256 threads per edge.
// ---------------------------------------------------------------------------
__global__ void agg_kernel(const float* __restrict__ sc,
                           const float* __restrict__ denom,
                           const float* __restrict__ v,
                           const int* __restrict__ ei,
                           float* __restrict__ agg,
                           int E) {
    const int e = blockIdx.x;
    const int d = threadIdx.x;          // 0..255
    const int h = d >> 5;
    const int tgt = ei[E + e];
    const float w = sc[(long long)e * NH + h] / denom[(long long)tgt * NH + h];
    atomicAdd(&agg[(long long)tgt * HD + d], w * v[(long long)tgt * HD + d]);
}

// ---------------------------------------------------------------------------
// Residual + LayerNorm: out = gamma*(x-mu)/sqrt(var+eps)+beta,  x = nodes+tmp
// One 128-thread block per node row, LDS tree reduction.
// ---------------------------------------------------------------------------
__global__ void ln_kernel(const float* __restrict__ nodes,
                          const float* __restrict__ tmp,
                          const float* __restrict__ gamma,
                          const float* __restrict__ beta,
                          float* __restrict__ out) {
    const int row = blockIdx.x;
    const int j = threadIdx.x;          // 0..127
    __shared__ float sbuf[ND];
    const long long base = (long long)row * ND + j;
    const float x = nodes[base] + tmp[base];

    sbuf[j] = x;
    __syncthreads();
#pragma unroll
    for (int s = ND / 2; s > 0; s >>= 1) {
        if (j < s) sbuf[j] += sbuf[j + s];
        __syncthreads();
    }
    const float mu = sbuf[0] * (1.0f / ND);
    __syncthreads();

    const float dxy = x - mu;
    sbuf[j] = dxy * dxy;
    __syncthreads();
#pragma unroll
    for (int s = ND / 2; s > 0; s >>= 1) {
        if (j < s) sbuf[j] += sbuf[j + s];
        __syncthreads();
    }
    const float var = sbuf[0] * (1.0f / ND);

    out[base] = gamma[j] * dxy * rsqrtf(var + LN_EPS) + beta[j];
}

// ---------------------------------------------------------------------------
// Host orchestration
// ---------------------------------------------------------------------------
extern "C" void kernel_launch(void* const* d_in, const int* in_sizes, int n_in,
                              void* d_out, int out_size, void* d_ws, size_t ws_size,
                              hipStream_t stream) {
    (void)in_sizes; (void)n_in; (void)out_size; (void)ws_size;

    static const int NN[4] = {40000, 10000, 2000, 400};
    static const int NE[4] = {320000, 160000, 32000, 6400};
    const int MAXN = 40000, MAXE = 320000;

    const float* Wq    = (const float*)d_in[12];
    const float* bq    = (const float*)d_in[13];
    const float* Wk    = (const float*)d_in[14];
    const float* bk    = (const float*)d_in[15];
    const float* Wv    = (const float*)d_in[16];
    const float* bv    = (const float*)d_in[17];
    const float* We    = (const float*)d_in[18];
    const float* be    = (const float*)d_in[19];
    const float* Wo    = (const float*)d_in[20];
    const float* bo    = (const float*)d_in[21];
    const float* gamma = (const float*)d_in[22];
    const float* beta  = (const float*)d_in[23];

    // Workspace layout (bytes), buffers reused across pipeline stages:
    //   bufA: q  -> v  -> tmp   (MAXN*HD floats)
    //   bufB: k  -> agg          (MAXN*HD floats)
    //   bufS: edge_att -> scores -> ex  (MAXE*NH floats)
    //   bufM: encoded segment max      (MAXN*NH uints)
    //   bufD: softmax denominators     (MAXN*NH floats)
    char* ws = (char*)d_ws;
    float*    bufA = (float*)(ws);
    float*    bufB = (float*)(ws + (size_t)MAXN * HD * 4);
    float*    bufS = (float*)(ws + (size_t)MAXN * HD * 4 * 2);
    unsigned* bufM = (unsigned*)(ws + (size_t)MAXN * HD * 4 * 2 + (size_t)MAXE * NH * 4);
    float*    bufD = (float*)(ws + (size_t)MAXN * HD * 4 * 2 + (size_t)MAXE * NH * 4
                                 + (size_t)MAXN * NH * 4);

    long long outOff = 0;
    for (int i = 0; i < 4; ++i) {
        const int Nn = NN[i];
        const int Ee = NE[i];
        const float* nodes = (const float*)d_in[3 * i + 0];
        const float* edges = (const float*)d_in[3 * i + 1];
        const int*   ei    = (const int*)  d_in[3 * i + 2];

        const dim3 blkG(128);
        const dim3 grdQKV((Nn + 15) / 16, (HD + 63) / 64);   // N=256 -> y=4
        const dim3 grdE  ((Ee + 15) / 16, (NH + 63) / 64);   // N=8   -> y=1
        const dim3 grdO  ((Nn + 15) / 16, (ND + 63) / 64);   // N=128 -> y=2

        // q, k projections
        wmma_gemm_bias<<<grdQKV, blkG, 0, stream>>>(nodes, Wq + (long long)i * ND * HD,
                                                    bq + i * HD, bufA, Nn, ND, HD);
        wmma_gemm_bias<<<grdQKV, blkG, 0, stream>>>(nodes, Wk + (long long)i * ND * HD,
                                                    bk + i * HD, bufB, Nn, ND, HD);
        // edge bias
        wmma_gemm_bias<<<grdE, blkG, 0, stream>>>(edges, We + (long long)i * ED * NH,
                                                  be + i * NH, bufS, Ee, ED, NH);

        // clear softmax stats
        {
            const int n = Nn * NH;
            fill_u32<<<(n + 255) / 256, 256, 0, stream>>>(bufM, 0u, n);
            fill_u32<<<(n + 255) / 256, 256, 0, stream>>>((unsigned*)bufD, 0u, n);
        }

        // scores + segment max, then exp + denominators
        {
            const int tot = Ee * NH;
            scores_kernel<<<(tot + 255) / 256, 256, 0, stream>>>(bufA, bufB, ei, bufS, bufM, Ee);
            exp_kernel<<<(tot + 255) / 256, 256, 0, stream>>>(bufS, bufM, bufD, ei, Ee);
        }

        // v projection (q dead -> reuse bufA)
        wmma_gemm_bias<<<grdQKV, blkG, 0, stream>>>(nodes, Wv + (long long)i * ND * HD,
                                                    bv + i * HD, bufA, Nn, ND, HD);

        // agg (k dead -> reuse bufB), clear first
        {
            const int n = Nn * HD;
            fill_u32<<<(n + 255) / 256, 256, 0, stream>>>((unsigned*)bufB, 0u, n);
        }
        agg_kernel<<<Ee, 256, 0, stream>>>(bufS, bufD, bufA, ei, bufB, Ee);

        // output projection (v dead -> tmp into bufA)
        wmma_gemm_bias<<<grdO, blkG, 0, stream>>>(bufB, Wo + (long long)i * HD * ND,
                                                  bo + i * ND, bufA, Nn, HD, ND);

        // residual + LayerNorm -> d_out slice
        ln_kernel<<<Nn, ND, 0, stream>>>(nodes, bufA, gamma + i * ND, beta + i * ND,
                                         (float*)d_out + outOff);
        outOff += (long long)Nn * ND;
    }
}